// Attention3Conv_10797547782217
// MI455X (gfx1250) — compile-verified
//
#include <hip/hip_runtime.h>
#include <math.h>

#define NNODES 100000
#define NEDGES 1600000
#define NFEAT  128
#define NHID   64
#define NGRAPH 64
#define BN_EPS 1e-5f

typedef __attribute__((ext_vector_type(2))) float v2f;
typedef __attribute__((ext_vector_type(8))) float v8f;

// ---------------------------------------------------------------- utilities
__global__ void fill_kernel(float* __restrict__ p, float v, int n) {
  int i = blockIdx.x * blockDim.x + threadIdx.x;
  if (i < n) p[i] = v;
}

__global__ void degree_kernel(const int* __restrict__ dst, float* __restrict__ deg) {
  int e = blockIdx.x * blockDim.x + threadIdx.x;
  if (e < NEDGES) unsafeAtomicAdd(&deg[dst[e]], 1.0f);   // global_atomic_add_f32
}

__global__ void rsqrt_kernel(float* __restrict__ d) {
  int i = blockIdx.x * blockDim.x + threadIdx.x;
  if (i < NNODES) d[i] = rsqrtf(d[i]);
}

// ------------------------------------------------- WMMA GEMM: out = A @ W
// A: [N x K] row-major, W: [K x 64] row-major, out: [N x 64].
// One wave computes one 16x16 tile with V_WMMA_F32_16X16X4_F32.
// Block = 256 threads = 8 waves -> 32 rows x 64 cols per block.
// f32 A 16x4 layout: lane m=lane&15; lanes<16 hold K={0,1}, lanes>=16 K={2,3}.
// f32 B 4x16 layout: lane n=lane&15; same K-pair split.
// f32 C/D 16x16: VGPR v holds M = v + 8*(lane>>4), N = lane&15.
__global__ void wmma_gemm_kernel(const float* __restrict__ A,
                                 const float* __restrict__ W,
                                 float* __restrict__ out, int K) {
  const int lane = threadIdx.x & 31;
  const int wave = threadIdx.x >> 5;
  const int row0 = blockIdx.x * 32 + (wave & 1) * 16;
  const int col0 = (wave >> 1) * 16;          // 4 column tiles cover H=64
  const int m    = lane & 15;                 // M for A, N for B
  const int kp   = (lane >> 4) * 2;           // K-pair select: 0 or 2
  const float* __restrict__ arow = A + (size_t)(row0 + m) * K;

  v8f acc = {};
  #pragma unroll 4
  for (int k0 = 0; k0 < K; k0 += 4) {
    __builtin_prefetch(arow + k0 + 32, 0, 0); // global_prefetch_b8 ahead in K
    v2f a, b;
    a.x = arow[k0 + kp];
    a.y = arow[k0 + kp + 1];
    b.x = W[(size_t)(k0 + kp)     * NHID + col0 + m];
    b.y = W[(size_t)(k0 + kp + 1) * NHID + col0 + m];
    // 8 args: (neg_a, A, neg_b, B, c_mod, C, reuse_a, reuse_b)
    acc = __builtin_amdgcn_wmma_f32_16x16x4_f32(false, a, false, b,
                                                (short)0, acc, false, false);
  }

  const int hi  = lane >> 4;
  const int col = col0 + m;
  #pragma unroll
  for (int v = 0; v < 8; ++v) {
    int row = row0 + v + 8 * hi;
    out[(size_t)row * NHID + col] = acc[v];
  }
}

// ------------------------------------------ edge scatter: agg[dst] += h[src]*norm
// One wave per edge; 2 columns per lane (H=64 / 32 lanes).
__global__ void edge_agg_kernel(const float* __restrict__ h,
                                const int* __restrict__ src,
                                const int* __restrict__ dst,
                                const float* __restrict__ dinv,
                                float* __restrict__ agg) {
  int gid  = blockIdx.x * blockDim.x + threadIdx.x;
  int e    = gid >> 5;
  int lane = gid & 31;
  if (e >= NEDGES) return;
  int s = src[e], d = dst[e];
  float w = dinv[s] * dinv[d];
  const float* __restrict__ hs = h + (size_t)s * NHID;
  float* __restrict__ ad = agg + (size_t)d * NHID;
  unsafeAtomicAdd(&ad[lane],      hs[lane]      * w);
  unsafeAtomicAdd(&ad[lane + 32], hs[lane + 32] * w);
}

// ---------------- combine self-loop + bias, accumulate BN column statistics
// Block handles 128 rows x 64 cols; hierarchical reduction: thread -> LDS ->
// one atomic per column per block (64 atomics instead of 8192).
__global__ void combine_stats_kernel(const float* __restrict__ h,
                                     float* __restrict__ agg,   // in: agg, out: combined
                                     const float* __restrict__ dinv,
                                     const float* __restrict__ bias,
                                     float* __restrict__ colsum,
                                     float* __restrict__ colsq) {
  __shared__ float s1[256];
  __shared__ float s2[256];
  int tid = threadIdx.x;
  int col = tid & 63;
  int rl  = tid >> 6;
  int base = blockIdx.x * 128;
  float sum = 0.f, sq = 0.f;
  for (int r = rl; r < 128; r += 4) {
    int n = base + r;
    if (n < NNODES) {
      size_t idx = (size_t)n * NHID + col;
      float di = dinv[n];
      float v  = agg[idx] + h[idx] * di * di + bias[col];
      agg[idx] = v;
      sum += v;
      sq  += v * v;
    }
  }
  s1[tid] = sum; s2[tid] = sq;
  __syncthreads();
  if (tid < 64) {
    float ts = s1[tid] + s1[tid + 64] + s1[tid + 128] + s1[tid + 192];
    float tq = s2[tid] + s2[tid + 64] + s2[tid + 128] + s2[tid + 192];
    unsafeAtomicAdd(&colsum[tid], ts);
    unsafeAtomicAdd(&colsq[tid],  tq);
  }
}

__global__ void bn_finalize_kernel(const float* __restrict__ colsum,
                                   const float* __restrict__ colsq,
                                   float* __restrict__ mean,
                                   float* __restrict__ invstd) {
  int j = threadIdx.x;
  if (j < NHID) {
    float mu  = colsum[j] * (1.0f / (float)NNODES);
    float var = colsq[j]  * (1.0f / (float)NNODES) - mu * mu;
    mean[j]   = mu;
    invstd[j] = rsqrtf(var + BN_EPS);
  }
}

__global__ void bn_relu_kernel(const float* __restrict__ in,
                               const float* __restrict__ mean,
                               const float* __restrict__ invstd,
                               const float* __restrict__ gamma,
                               const float* __restrict__ beta,
                               float* __restrict__ out) {
  size_t idx = (size_t)blockIdx.x * blockDim.x + threadIdx.x;
  if (idx < (size_t)NNODES * NHID) {
    int col = (int)(idx & 63);
    float v = (in[idx] - mean[col]) * invstd[col] * gamma[col] + beta[col];
    out[idx] = v > 0.f ? v : 0.f;
  }
}

// --------------------- attention: att = sigmoid(h @ att_w); seg[batch] += h*att
__global__ void att_pool_kernel(const float* __restrict__ h,
                                const float* __restrict__ aw,
                                const int* __restrict__ batch,
                                float* __restrict__ att_out,
                                float* __restrict__ seg) {
  int gid  = blockIdx.x * blockDim.x + threadIdx.x;
  int n    = gid >> 5;
  int lane = gid & 31;
  if (n >= NNODES) return;
  const float* __restrict__ hn = h + (size_t)n * NHID;
  float h0 = hn[lane], h1 = hn[lane + 32];
  float p = h0 * aw[lane] + h1 * aw[lane + 32];
  #pragma unroll
  for (int off = 16; off > 0; off >>= 1) p += __shfl_xor(p, off, 32);
  float att = 1.0f / (1.0f + __expf(-p));
  if (lane == 0) att_out[n] = att;
  float* __restrict__ sg = seg + (size_t)batch[n] * NHID;
  unsafeAtomicAdd(&sg[lane],      h0 * att);
  unsafeAtomicAdd(&sg[lane + 32], h1 * att);
}

__global__ void logits_kernel(const float* __restrict__ seg,
                              const float* __restrict__ cw,
                              const float* __restrict__ cb,
                              float* __restrict__ out) {
  int gid  = blockIdx.x * blockDim.x + threadIdx.x;
  int g    = gid >> 5;
  int lane = gid & 31;
  if (g >= NGRAPH) return;
  const float* __restrict__ sg = seg + (size_t)g * NHID;
  float p = sg[lane] * cw[lane] + sg[lane + 32] * cw[lane + 32];
  #pragma unroll
  for (int off = 16; off > 0; off >>= 1) p += __shfl_xor(p, off, 32);
  if (lane == 0) out[g] = p + cb[0];
}

// ---------------------------------------------------------------- launcher
extern "C" void kernel_launch(void* const* d_in, const int* in_sizes, int n_in,
                              void* d_out, int out_size, void* d_ws, size_t ws_size,
                              hipStream_t stream) {
  const float* x     = (const float*)d_in[0];
  const int*   ei    = (const int*)  d_in[1];
  const int*   batch = (const int*)  d_in[2];
  const float* W1 = (const float*)d_in[3];  const float* b1  = (const float*)d_in[4];
  const float* g1 = (const float*)d_in[5];  const float* be1 = (const float*)d_in[6];
  const float* W2 = (const float*)d_in[7];  const float* b2  = (const float*)d_in[8];
  const float* g2 = (const float*)d_in[9];  const float* be2 = (const float*)d_in[10];
  const float* W3 = (const float*)d_in[11]; const float* b3  = (const float*)d_in[12];
  const float* g3 = (const float*)d_in[13]; const float* be3 = (const float*)d_in[14];
  const float* aw = (const float*)d_in[15];
  const float* cw = (const float*)d_in[16]; const float* cb  = (const float*)d_in[17];

  const int* src = ei;            // edge_index[0]
  const int* dst = ei + NEDGES;   // edge_index[1]

  // workspace carve-up (floats)
  float* dinv  = (float*)d_ws;                       // N
  float* hA    = dinv + NNODES;                      // N*H  (gemm output)
  float* hB    = hA + (size_t)NNODES * NHID;         // N*H  (layer output)
  float* hC    = hB + (size_t)NNODES * NHID;         // N*H  (agg / combined)
  float* stats = hC + (size_t)NNODES * NHID;         // 4*H: colsum, colsq, mean, invstd
  float* seg   = stats + 4 * NHID;                   // G*H

  float* logits = (float*)d_out;                     // [G]
  float* att    = logits + NGRAPH;                   // [N]

  const int elemNH = NNODES * NHID;

  // symmetric-normalization degrees (self-loop included: start at 1.0)
  fill_kernel<<<(NNODES + 255) / 256, 256, 0, stream>>>(dinv, 1.0f, NNODES);
  degree_kernel<<<(NEDGES + 255) / 256, 256, 0, stream>>>(dst, dinv);
  rsqrt_kernel<<<(NNODES + 255) / 256, 256, 0, stream>>>(dinv);

  auto layer = [&](const float* in, const float* W, const float* b,
                   const float* gamma, const float* beta, float* outp, int K) {
    wmma_gemm_kernel<<<NNODES / 32, 256, 0, stream>>>(in, W, hA, K);
    fill_kernel<<<(elemNH + 255) / 256, 256, 0, stream>>>(hC, 0.0f, elemNH);
    fill_kernel<<<1, 256, 0, stream>>>(stats, 0.0f, 2 * NHID);
    edge_agg_kernel<<<(NEDGES * 32) / 256, 256, 0, stream>>>(hA, src, dst, dinv, hC);
    combine_stats_kernel<<<(NNODES + 127) / 128, 256, 0, stream>>>(
        hA, hC, dinv, b, stats, stats + NHID);
    bn_finalize_kernel<<<1, 64, 0, stream>>>(stats, stats + NHID,
                                             stats + 2 * NHID, stats + 3 * NHID);
    bn_relu_kernel<<<(elemNH + 255) / 256, 256, 0, stream>>>(
        hC, stats + 2 * NHID, stats + 3 * NHID, gamma, beta, outp);
  };

  layer(x,  W1, b1, g1, be1, hB, NFEAT);
  layer(hB, W2, b2, g2, be2, hB, NHID);
  layer(hB, W3, b3, g3, be3, hB, NHID);

  fill_kernel<<<(NGRAPH * NHID + 255) / 256, 256, 0, stream>>>(seg, 0.0f, NGRAPH * NHID);
  att_pool_kernel<<<(NNODES * 32) / 256, 256, 0, stream>>>(hB, aw, batch, att, seg);
  logits_kernel<<<(NGRAPH * 32 + 255) / 256, 256, 0, stream>>>(seg, cw, cb, logits);
}